// Group_3599182594916
// MI455X (gfx1250) — compile-verified
//
#include <hip/hip_runtime.h>
#include <hip/hip_bf16.h>

// Problem constants (xyz: [B, N, 3] f32)
#define BATCH     16
#define NPTS      8192
#define NGROUP    512
#define KNN_M     32
#define CHUNK     512                 // score columns buffered in LDS per pass
#define NCHUNK    (NPTS / CHUNK)      // 16
#define TILES_PER_CHUNK (CHUNK / 16)  // 32
#define TILES_PER_WAVE  (TILES_PER_CHUNK / 8)  // 4

// d_out flat layout (floats), reference return order:
//   neighborhood [B,G,M,3] : 786432   @ 0
//   center       [B,G,3]   : 24576    @ 786432
//   idx_flat     [B*G*M]   : 262144   @ 811008
//   center_idx   [B*G]     : 8192     @ 1073152
#define OFF_CENTER   786432
#define OFF_IDX      811008
#define OFF_CIDX     1073152

typedef __attribute__((ext_vector_type(2))) float v2f;
typedef __attribute__((ext_vector_type(8))) float v8f;

// ---------------------------------------------------------------------------
// Kernel 1: Farthest Point Sampling. One 1024-thread WG per batch.
// Points live in registers (8/thread); per-iteration argmax via wave shfl
// reduction + one 32-entry LDS reduction by wave 0. Writes center coords and
// center_idx (as float) to d_out.
// ---------------------------------------------------------------------------
__global__ __launch_bounds__(1024) void fps_kernel(const float* __restrict__ xyz,
                                                   float* __restrict__ out) {
    __shared__ float sp[3];
    __shared__ int   sLast;
    __shared__ float wv[32];
    __shared__ int   wi[32];

    const int b    = blockIdx.x;
    const int tid  = threadIdx.x;
    const int lane = tid & 31;
    const int wave = tid >> 5;

    float px_[8], py_[8], pz_[8], d_[8];
#pragma unroll
    for (int k = 0; k < 8; ++k) {
        const int i = tid + (k << 10);
        const float* p = xyz + ((size_t)b * NPTS + i) * 3;
        px_[k] = p[0]; py_[k] = p[1]; pz_[k] = p[2];
        d_[k]  = 1.0e10f;                       // BIG from reference
    }

    float* centerOut = out + OFF_CENTER + (size_t)b * NGROUP * 3;
    float* cidxOut   = out + OFF_CIDX   + (size_t)b * NGROUP;

    if (tid == 0) {                              // first pick: index 0
        sp[0] = px_[0]; sp[1] = py_[0]; sp[2] = pz_[0];
        centerOut[0] = px_[0]; centerOut[1] = py_[0]; centerOut[2] = pz_[0];
        cidxOut[0] = (float)((size_t)b * NPTS);
    }
    __syncthreads();

    for (int it = 1; it < NGROUP; ++it) {
        const float px = sp[0], py = sp[1], pz = sp[2];
        float bv = -1.0f; int bi = 0x7fffffff;
#pragma unroll
        for (int k = 0; k < 8; ++k) {
            const float dx = px_[k] - px, dy = py_[k] - py, dz = pz_[k] - pz;
            const float dd = dx * dx + dy * dy + dz * dz;
            d_[k] = fminf(d_[k], dd);
            const int i = tid + (k << 10);
            if (d_[k] > bv || (d_[k] == bv && i < bi)) { bv = d_[k]; bi = i; }
        }
        // wave32 argmax, tie-break lower index (matches jnp.argmax)
#pragma unroll
        for (int off = 1; off < 32; off <<= 1) {
            const float ov = __shfl_xor(bv, off, 32);
            const int   oi = __shfl_xor(bi, off, 32);
            if (ov > bv || (ov == bv && oi < bi)) { bv = ov; bi = oi; }
        }
        if (lane == 0) { wv[wave] = bv; wi[wave] = bi; }
        __syncthreads();
        if (wave == 0) {
            bv = wv[lane]; bi = wi[lane];
#pragma unroll
            for (int off = 1; off < 32; off <<= 1) {
                const float ov = __shfl_xor(bv, off, 32);
                const int   oi = __shfl_xor(bi, off, 32);
                if (ov > bv || (ov == bv && oi < bi)) { bv = ov; bi = oi; }
            }
            if (lane == 0) sLast = bi;
        }
        __syncthreads();
        const int last = sLast;
        if (tid == (last & 1023)) {              // owner broadcasts its coords
            const int k = last >> 10;
            sp[0] = px_[k]; sp[1] = py_[k]; sp[2] = pz_[k];
            centerOut[it * 3 + 0] = px_[k];
            centerOut[it * 3 + 1] = py_[k];
            centerOut[it * 3 + 2] = pz_[k];
            cidxOut[it] = (float)(last + (size_t)b * NPTS);
        }
        __syncthreads();
    }
}

// ---------------------------------------------------------------------------
// Kernel 2: KNN via V_WMMA_F32_16X16X4_F32 + fused top-32 selection.
// Grid: B * (G/16) = 512 WGs, 256 threads (8 waves).
// A (16x4): rows = 16 centers, cols = (-2x, -2y, -2z, 1)   (scale folded here)
// B (4x16): rows = (rx, ry, rz, |r|^2), cols = 16 candidate points
// C = A*B = -2 q.r + |r|^2  (ordering-equivalent to squared distance)
// Fragments are built branchlessly (v_cndmask, no EXEC branches) and the
// 4 tiles/wave/chunk are software-pipelined so point loads overlap WMMA.
// Scores stream through a 32KB LDS chunk; 16 threads/row keep register top-32
// lists, merged at the end with a 16-lane shfl k-way merge (sorted, tie-break
// lower index, matching lax.top_k).
// ---------------------------------------------------------------------------
__global__ __launch_bounds__(256) void knn_group_kernel(const float* __restrict__ xyz,
                                                        float* __restrict__ out) {
    __shared__ float scores[16 * CHUNK];         // 32 KB

    const int b    = blockIdx.x >> 5;            // 32 group-tiles per batch
    const int g0   = (blockIdx.x & 31) << 4;
    const int tid  = threadIdx.x;
    const int lane = tid & 31;
    const int wave = tid >> 5;
    const bool lo  = (lane < 16);

    const float* centerBase = out + OFF_CENTER;

    // ---- A fragment (built once, branchless) ----------------------------
    // 32-bit A 16x4 layout: lanes 0-15 hold K=0,1; lanes 16-31 hold K=2,3
    const int m = lane & 15;
    const float* q = centerBase + ((size_t)(b * NGROUP + g0 + m)) * 3;
    const float qx = q[0], qy = q[1], qz = q[2];
    v2f afrag;
    afrag.x = lo ? (-2.0f * qx) : (-2.0f * qz);
    afrag.y = lo ? (-2.0f * qy) : 1.0f;

    // ---- per-thread sorted top-32 over a 512-column slice ---------------
    float Ld[KNN_M]; int Li[KNN_M];
#pragma unroll
    for (int i = 0; i < KNN_M; ++i) { Ld[i] = 3.0e38f; Li[i] = 0x7fffffff; }
    const int selRow = tid >> 4;                 // 16 threads per query row
    const int selT   = tid & 15;                 // contiguous lanes -> no LDS conflicts

    const int rbase = lo ? 0 : 8;                // C rows this lane holds

    for (int c = 0; c < NCHUNK; ++c) {
        // -------- compute phase: 32 tiles / chunk, 4 per wave, pipelined --
        const float* rp = xyz + ((size_t)b * NPTS + c * CHUNK + wave * 16 + (lane & 15)) * 3;
        float rx = rp[0], ry = rp[1], rz = rp[2];
#pragma unroll
        for (int tw = 0; tw < TILES_PER_WAVE; ++tw) {
            // prefetch next tile's point before the WMMA (overlap latency)
            float nx = 0.0f, ny = 0.0f, nz = 0.0f;
            if (tw + 1 < TILES_PER_WAVE) {
                const float* np = rp + (8 * 16) * 3;   // tile stride = 8 tiles
                nx = np[0]; ny = np[1]; nz = np[2];
                rp = np;
            }

            const float nrm = rx * rx + ry * ry + rz * rz;
            v2f bfrag;                            // branchless select
            bfrag.x = lo ? rx : rz;
            bfrag.y = lo ? ry : nrm;

            v8f cacc = {};
            cacc = __builtin_amdgcn_wmma_f32_16x16x4_f32(
                false, afrag, false, bfrag, (short)0, cacc, false, false);

            // C layout: VGPR v -> row (v | v+8), col = lane&15 within tile
            const int tile = wave + 8 * tw;
            const int cic  = tile * 16 + (lane & 15);
#pragma unroll
            for (int v = 0; v < 8; ++v)
                scores[(rbase + v) * CHUNK + cic] = cacc[v];

            rx = nx; ry = ny; rz = nz;
        }
        __syncthreads();

        // -------- selection phase: scan 32 columns of this chunk ---------
        for (int j = 0; j < CHUNK / 16; ++j) {
            const int   cc = selT + j * 16;
            const float v  = scores[selRow * CHUNK + cc];
            if (v < Ld[KNN_M - 1]) {
                const int gcol = c * CHUNK + cc;
                int p = KNN_M - 1;
                while (p > 0 && Ld[p - 1] > v) { Ld[p] = Ld[p - 1]; Li[p] = Li[p - 1]; --p; }
                Ld[p] = v; Li[p] = gcol;
            }
        }
        __syncthreads();
    }

    // ---- merge: 16 sorted lists per row -> global top-32 (sorted) -------
    const int g = g0 + selRow;
    const float cx = centerBase[(size_t)(b * NGROUP + g) * 3 + 0];
    const float cy = centerBase[(size_t)(b * NGROUP + g) * 3 + 1];
    const float cz = centerBase[(size_t)(b * NGROUP + g) * 3 + 2];
    float* nbrOut = out + ((size_t)(b * NGROUP + g)) * KNN_M * 3;
    float* idxOut = out + OFF_IDX + ((size_t)(b * NGROUP + g)) * KNN_M;

    int ptr = 0;
    for (int o = 0; o < KNN_M; ++o) {
        float v  = (ptr < KNN_M) ? Ld[ptr] : 3.0e38f;
        int   ci = (ptr < KNN_M) ? Li[ptr] : 0x7fffffff;
#pragma unroll
        for (int off = 1; off < 16; off <<= 1) {  // half-wave min-reduce
            const float ov = __shfl_xor(v,  off, 32);
            const int   oc = __shfl_xor(ci, off, 32);
            if (ov < v || (ov == v && oc < ci)) { v = ov; ci = oc; }
        }
        if (ptr < KNN_M && Li[ptr] == ci) ++ptr;  // unique cols -> one winner
        if (selT == (o & 15)) {                   // spread gather/stores
            const float* r = xyz + ((size_t)b * NPTS + ci) * 3;
            nbrOut[o * 3 + 0] = r[0] - cx;
            nbrOut[o * 3 + 1] = r[1] - cy;
            nbrOut[o * 3 + 2] = r[2] - cz;
            idxOut[o] = (float)(ci + (size_t)b * NPTS);
        }
    }
}

// ---------------------------------------------------------------------------
extern "C" void kernel_launch(void* const* d_in, const int* in_sizes, int n_in,
                              void* d_out, int out_size, void* d_ws, size_t ws_size,
                              hipStream_t stream) {
    const float* xyz = (const float*)d_in[0];
    float* out = (float*)d_out;

    fps_kernel<<<BATCH, 1024, 0, stream>>>(xyz, out);
    knn_group_kernel<<<BATCH * (NGROUP / 16), 256, 0, stream>>>(xyz, out);
}